// TypeEnhancedGCN_10539849744794
// MI455X (gfx1250) — compile-verified
//
#include <hip/hip_runtime.h>

// Problem constants (from reference)
#define BB 8
#define SS 128
#define HID 768
#define DEP 64
#define DSZ 768
#define NT 45
#define NEGV -1000000000.0f

typedef __attribute__((ext_vector_type(2))) float v2f;
typedef __attribute__((ext_vector_type(8))) float v8f;

// ---------------------------------------------------------------------------
// Batched NN GEMM: C[M,N] = A[M,K] @ B[K,N] (+bias) (opt relu)
// block = 128 threads = 4 waves. Each wave computes a 16x64 output strip:
// 4 accumulators sharing one A fragment per K-step (4x A reuse), pointer-
// stepped loads so B accesses become immediate-offset global_loads.
// Matrix op: V_WMMA_F32_16X16X4_F32 (full f32, matches f32 reference).
// ---------------------------------------------------------------------------
__global__ void wmma_gemm_nn(const float* __restrict__ A,
                             const float* __restrict__ B,
                             const float* __restrict__ bias,
                             float* __restrict__ C,
                             int M, int N, int K,
                             long long sA, long long sB, long long sC,
                             int relu) {
    const int wave = threadIdx.x >> 5;
    const int lane = threadIdx.x & 31;
    const int stripsN   = N >> 6;                 // strips of 64 columns
    const int numStrips = (M >> 4) * stripsN;
    const int stripId = blockIdx.x * 4 + wave;
    if (stripId >= numStrips) return;             // wave-uniform: EXEC stays all-1s
    const int tm = stripId / stripsN;
    const int sn = stripId % stripsN;

    const float* Ab = A + (long long)blockIdx.y * sA;
    const float* Bb = B + (long long)blockIdx.y * sB;
    float*       Cb = C + (long long)blockIdx.y * sC;

    const int m    = lane & 15;                   // A row in tile / B,C col in tile
    const int koff = (lane >> 4) << 1;            // lanes 0-15: K 0,1 ; 16-31: K 2,3
    const int col0 = (sn << 6) + m;               // first of 4 column groups

    const float* aptr = Ab + (long long)((tm << 4) + m) * K + koff;
    const float* bptr = Bb + (long long)koff * N + col0;

    v8f acc0 = {}, acc1 = {}, acc2 = {}, acc3 = {};
    for (int k0 = 0; k0 < K; k0 += 4) {
        v2f a = *(const v2f*)aptr;                // 8B-aligned (K, koff even)
        v2f b0, b1, b2, b3;
        b0.x = bptr[0];      b0.y = bptr[N];
        b1.x = bptr[16];     b1.y = bptr[N + 16];
        b2.x = bptr[32];     b2.y = bptr[N + 32];
        b3.x = bptr[48];     b3.y = bptr[N + 48];
        acc0 = __builtin_amdgcn_wmma_f32_16x16x4_f32(false, a, false, b0, (short)0, acc0, false, false);
        acc1 = __builtin_amdgcn_wmma_f32_16x16x4_f32(false, a, false, b1, (short)0, acc1, false, false);
        acc2 = __builtin_amdgcn_wmma_f32_16x16x4_f32(false, a, false, b2, (short)0, acc2, false, false);
        acc3 = __builtin_amdgcn_wmma_f32_16x16x4_f32(false, a, false, b3, (short)0, acc3, false, false);
        aptr += 4;
        bptr += (long long)4 * N;
    }

    const int rbase = (tm << 4) + ((lane >> 4) << 3);   // M = r + 8*(lane>=16)
    v8f accs[4] = {acc0, acc1, acc2, acc3};
    #pragma unroll
    for (int j = 0; j < 4; ++j) {
        const int col = col0 + 16 * j;
        const float bv = bias ? bias[col] : 0.0f;
        #pragma unroll
        for (int r = 0; r < 8; ++r) {
            float v = accs[j][r] + bv;
            if (relu) v = fmaxf(v, 0.0f);
            Cb[(long long)(rbase + r) * N + col] = v;
        }
    }
}

// ---------------------------------------------------------------------------
// Batched NT GEMM: C[M,N] = A[M,K] @ B[N,K]^T   (Gram term G = u1 u2^T)
// Same 16x64-strip structure; B fragments are contiguous v2f along K.
// ---------------------------------------------------------------------------
__global__ void wmma_gemm_nt(const float* __restrict__ A,
                             const float* __restrict__ B,
                             float* __restrict__ C,
                             int M, int N, int K,
                             long long sA, long long sB, long long sC) {
    const int wave = threadIdx.x >> 5;
    const int lane = threadIdx.x & 31;
    const int stripsN   = N >> 6;
    const int numStrips = (M >> 4) * stripsN;
    const int stripId = blockIdx.x * 4 + wave;
    if (stripId >= numStrips) return;
    const int tm = stripId / stripsN;
    const int sn = stripId % stripsN;

    const float* Ab = A + (long long)blockIdx.y * sA;
    const float* Bb = B + (long long)blockIdx.y * sB;
    float*       Cb = C + (long long)blockIdx.y * sC;

    const int m    = lane & 15;
    const int koff = (lane >> 4) << 1;
    const int col0 = (sn << 6) + m;

    const float* aptr = Ab + (long long)((tm << 4) + m) * K + koff;
    const float* bptr = Bb + (long long)col0 * K + koff;   // row col0 of B
    const long long bstride16 = (long long)16 * K;         // 16 rows ahead

    v8f acc0 = {}, acc1 = {}, acc2 = {}, acc3 = {};
    for (int k0 = 0; k0 < K; k0 += 4) {
        v2f a  = *(const v2f*)aptr;
        v2f b0 = *(const v2f*)(bptr);
        v2f b1 = *(const v2f*)(bptr + bstride16);
        v2f b2 = *(const v2f*)(bptr + 2 * bstride16);
        v2f b3 = *(const v2f*)(bptr + 3 * bstride16);
        acc0 = __builtin_amdgcn_wmma_f32_16x16x4_f32(false, a, false, b0, (short)0, acc0, false, false);
        acc1 = __builtin_amdgcn_wmma_f32_16x16x4_f32(false, a, false, b1, (short)0, acc1, false, false);
        acc2 = __builtin_amdgcn_wmma_f32_16x16x4_f32(false, a, false, b2, (short)0, acc2, false, false);
        acc3 = __builtin_amdgcn_wmma_f32_16x16x4_f32(false, a, false, b3, (short)0, acc3, false, false);
        aptr += 4;
        bptr += 4;
    }

    const int rbase = (tm << 4) + ((lane >> 4) << 3);
    v8f accs[4] = {acc0, acc1, acc2, acc3};
    #pragma unroll
    for (int j = 0; j < 4; ++j) {
        const int col = col0 + 16 * j;
        #pragma unroll
        for (int r = 0; r < 8; ++r)
            Cb[(long long)(rbase + r) * N + col] = accs[j][r];
    }
}

// ---------------------------------------------------------------------------
// T[45,768] = emb[45,64] @ W[64,768]   (tiny; scalar)
// ---------------------------------------------------------------------------
__global__ void emb_gemm(const float* __restrict__ E,
                         const float* __restrict__ W,
                         float* __restrict__ T) {
    int idx = blockIdx.x * blockDim.x + threadIdx.x;
    if (idx >= NT * DSZ) return;
    int r = idx / DSZ, n = idx % DSZ;
    float s = 0.0f;
    for (int k = 0; k < DEP; ++k) s += E[r * DEP + k] * W[(long long)k * DSZ + n];
    T[idx] = s;
}

// c[t] = dot(T1[t], T2[t]) over DSZ
__global__ void compute_c(const float* __restrict__ T1,
                          const float* __restrict__ T2,
                          float* __restrict__ cvec) {
    int t = threadIdx.x;
    if (t >= NT) return;
    float s = 0.0f;
    for (int e = 0; e < DSZ; ++e) s += T1[t * DSZ + e] * T2[t * DSZ + e];
    cvec[t] = s;
}

// P[row,t] = dot(U[row,:], T[t,:])   rows = B*S
__global__ void compute_P(const float* __restrict__ U,
                          const float* __restrict__ T,
                          float* __restrict__ P) {
    int idx = blockIdx.x * blockDim.x + threadIdx.x;
    if (idx >= BB * SS * NT) return;
    int row = idx / NT, t = idx % NT;
    float s = 0.0f;
    const float* u  = U + (long long)row * DSZ;
    const float* tv = T + (long long)t * DSZ;
    for (int e = 0; e < DSZ; ++e) s += u[e] * tv[e];
    P[idx] = s;
}

// ---------------------------------------------------------------------------
// relevance assembly + masked softmax over j.
// grid = (S, B), block = S(=128). rel = G + P1[b,i,t] + P2[b,j,t] + c[t].
// ---------------------------------------------------------------------------
__global__ void rel_softmax(const float* __restrict__ G,
                            const float* __restrict__ P1,
                            const float* __restrict__ P2,
                            const float* __restrict__ cvec,
                            const int* __restrict__ types,
                            float* __restrict__ Aout) {
    const int i = blockIdx.x, b = blockIdx.y, j = threadIdx.x;
    const long long rowOff = ((long long)b * SS + i) * SS;
    const int t = types[rowOff + j];

    float rel = G[rowOff + j]
              + P1[((long long)b * SS + i) * NT + t]
              + P2[((long long)b * SS + j) * NT + t]
              + cvec[t];
    if (t == 0) rel = NEGV;

    __shared__ float red[SS];
    red[j] = rel;
    __syncthreads();
    for (int s = SS / 2; s > 0; s >>= 1) {
        if (j < s) red[j] = fmaxf(red[j], red[j + s]);
        __syncthreads();
    }
    const float mx = red[0];
    __syncthreads();

    const float e = __expf(rel - mx);
    red[j] = e;
    __syncthreads();
    for (int s = SS / 2; s > 0; s >>= 1) {
        if (j < s) red[j] += red[j + s];
        __syncthreads();
    }
    Aout[rowOff + j] = e / red[0];
}

// ---------------------------------------------------------------------------
extern "C" void kernel_launch(void* const* d_in, const int* in_sizes, int n_in,
                              void* d_out, int out_size, void* d_ws, size_t ws_size,
                              hipStream_t stream) {
    const float* h     = (const float*)d_in[0];   // [B,S,HID]
    const int*   types = (const int*)d_in[1];     // [B,S,S]
    const float* emb   = (const float*)d_in[2];   // [NT,DEP]
    const float* W1w   = (const float*)d_in[3];   // [HID+DEP, DS]
    const float* W1b   = (const float*)d_in[4];
    const float* W2w   = (const float*)d_in[5];
    const float* W2b   = (const float*)d_in[6];
    const float* gcnw  = (const float*)d_in[7];   // [HID,HID]
    const float* gcnb  = (const float*)d_in[8];
    const float* Wow   = (const float*)d_in[9];   // [DS,HID]
    const float* Wob   = (const float*)d_in[10];
    float* out = (float*)d_out;

    // workspace carve-out
    char* ws = (char*)d_ws;
    size_t off = 0;
    auto carve = [&](size_t bytes) -> float* {
        float* p = (float*)(ws + off);
        off = (off + bytes + 255) & ~(size_t)255;
        return p;
    };
    const int ROWS = BB * SS;                         // 1024
    float* u1   = carve((size_t)ROWS * DSZ * 4);      // 3 MB
    float* u2   = carve((size_t)ROWS * DSZ * 4);      // 3 MB
    float* T1   = carve((size_t)NT * DSZ * 4);
    float* T2   = carve((size_t)NT * DSZ * 4);
    float* cvec = carve((size_t)NT * 4);
    float* P1   = carve((size_t)ROWS * NT * 4);
    float* P2   = carve((size_t)ROWS * NT * 4);
    float* G    = carve((size_t)BB * SS * SS * 4);    // 512 KB
    float* Amat = carve((size_t)BB * SS * SS * 4);    // 512 KB
    float* agg  = carve((size_t)ROWS * HID * 4);      // 3 MB
    float* hsyn = carve((size_t)ROWS * HID * 4);      // 3 MB

    const dim3 blk(128);

    // u1 = h @ W1[:HID] + b1 ; u2 = h @ W2[:HID] + b2     [1024,768]x[768,768]
    {
        int strips = (ROWS / 16) * (DSZ / 64);        // 64 * 12 = 768
        dim3 grid((strips + 3) / 4, 1);
        wmma_gemm_nn<<<grid, blk, 0, stream>>>(h, W1w, W1b, u1,
                                               ROWS, DSZ, HID, 0, 0, 0, 0);
        wmma_gemm_nn<<<grid, blk, 0, stream>>>(h, W2w, W2b, u2,
                                               ROWS, DSZ, HID, 0, 0, 0, 0);
    }

    // T1/T2 = emb @ W{1,2}[HID:] ; c[t] = T1[t]·T2[t]
    emb_gemm<<<(NT * DSZ + 255) / 256, 256, 0, stream>>>(emb, W1w + (size_t)HID * DSZ, T1);
    emb_gemm<<<(NT * DSZ + 255) / 256, 256, 0, stream>>>(emb, W2w + (size_t)HID * DSZ, T2);
    compute_c<<<1, 64, 0, stream>>>(T1, T2, cvec);

    // P1[row,t] = u1·T2 ; P2[row,t] = u2·T1
    compute_P<<<(ROWS * NT + 255) / 256, 256, 0, stream>>>(u1, T2, P1);
    compute_P<<<(ROWS * NT + 255) / 256, 256, 0, stream>>>(u2, T1, P2);

    // G[b] = u1[b] @ u2[b]^T   (per batch: [128,768]x[768,128])
    {
        int strips = (SS / 16) * (SS / 64);           // 8 * 2 = 16
        dim3 grid((strips + 3) / 4, BB);
        wmma_gemm_nt<<<grid, blk, 0, stream>>>(u1, u2, G, SS, SS, DSZ,
                                               (long long)SS * DSZ,
                                               (long long)SS * DSZ,
                                               (long long)SS * SS);
    }

    // masked softmax -> A
    rel_softmax<<<dim3(SS, BB), dim3(SS), 0, stream>>>(G, P1, P2, cvec, types, Amat);

    // agg[b] = A[b] @ h[b]     (per batch: [128,128]x[128,768])
    {
        int strips = (SS / 16) * (HID / 64);          // 8 * 12 = 96
        dim3 grid((strips + 3) / 4, BB);
        wmma_gemm_nn<<<grid, blk, 0, stream>>>(Amat, h, nullptr, agg,
                                               SS, HID, SS,
                                               (long long)SS * SS,
                                               (long long)SS * HID,
                                               (long long)SS * HID, 0);
    }

    // hsyn = relu(agg @ gcn_w + gcn_b) ; out = hsyn @ Wo_w + Wo_b
    {
        int strips = (ROWS / 16) * (HID / 64);
        dim3 grid((strips + 3) / 4, 1);
        wmma_gemm_nn<<<grid, blk, 0, stream>>>(agg, gcnw, gcnb, hsyn,
                                               ROWS, HID, HID, 0, 0, 0, 1);
        wmma_gemm_nn<<<grid, blk, 0, stream>>>(hsyn, Wow, Wob, out,
                                               ROWS, HID, DSZ, 0, 0, 0, 0);
    }
}